// LinearAndGather_78280073937657
// MI455X (gfx1250) — compile-verified
//
#include <hip/hip_runtime.h>

typedef __attribute__((ext_vector_type(2))) float v2f;
typedef __attribute__((ext_vector_type(4))) float f32x4;
typedef __attribute__((ext_vector_type(8))) float v8f;

#define DIM 64

// ---------------------------------------------------------------------------
// Kernel 1: z[n, o] = sum_i w[o, i] * values[n, i]   (z = values * W^T)
// One wave32 per 16-node tile. A = values tile (16x4 per step), B = W^T chunk
// (4x16), four f32 accumulators cover o = 0..63. Uses V_WMMA_F32_16X16X4_F32.
// ---------------------------------------------------------------------------
__global__ void __launch_bounds__(256)
linear_wmma_kernel(const float* __restrict__ values,
                   const float* __restrict__ w,
                   float* __restrict__ z,
                   int n_nodes, int n_tiles) {
    const int wave = threadIdx.x >> 5;
    const int lane = threadIdx.x & 31;
    const int tile = blockIdx.x * 8 + wave;
    if (tile >= n_tiles) return;          // wave-uniform: EXEC stays all-ones

    const int row_base = tile << 4;
    const int m    = lane & 15;           // A: row M ; B: col N ; D: col N
    const int half = lane >> 4;           // K half selector for A and B

    int arow = row_base + m;
    if (arow >= n_nodes) arow = n_nodes - 1;   // clamp loads; stores guarded
    const float* ap = values + (size_t)arow * DIM + 2 * half;
    const float* wp = w + (size_t)m * DIM + 2 * half;   // row m of W = col m of B

    v8f acc0 = {}, acc1 = {}, acc2 = {}, acc3 = {};

#pragma unroll
    for (int k = 0; k < DIM / 4; ++k) {
        const int i0 = k * 4;
        v2f a;                       // A[m, i0 + 2*half .. +1]
        a.x = ap[i0 + 0];
        a.y = ap[i0 + 1];
        v2f b0, b1, b2, b3;          // B[i0 + 2*half .. +1, o_group*16 + m]
        b0.x = wp[i0 + 0];              b0.y = wp[i0 + 1];
        b1.x = wp[16 * DIM + i0 + 0];   b1.y = wp[16 * DIM + i0 + 1];
        b2.x = wp[32 * DIM + i0 + 0];   b2.y = wp[32 * DIM + i0 + 1];
        b3.x = wp[48 * DIM + i0 + 0];   b3.y = wp[48 * DIM + i0 + 1];
        acc0 = __builtin_amdgcn_wmma_f32_16x16x4_f32(false, a, false, b0, (short)0, acc0, false, false);
        acc1 = __builtin_amdgcn_wmma_f32_16x16x4_f32(false, a, false, b1, (short)0, acc1, false, false);
        acc2 = __builtin_amdgcn_wmma_f32_16x16x4_f32(false, a, false, b2, (short)0, acc2, false, false);
        acc3 = __builtin_amdgcn_wmma_f32_16x16x4_f32(false, a, false, b3, (short)0, acc3, false, false);
    }

    // D layout: VGPR r -> M = r + 8*half, N = m
#pragma unroll
    for (int r = 0; r < 8; ++r) {
        const int mm  = r + half * 8;
        const int row = row_base + mm;
        if (row < n_nodes) {
            float* zp = z + (size_t)row * DIM + m;
            zp[0]  = acc0[r];
            zp[16] = acc1[r];
            zp[32] = acc2[r];
            zp[48] = acc3[r];
        }
    }
}

// ---------------------------------------------------------------------------
// Kernel 2: out[row, :] = z[idx_in[idx_out[row]], :]
// 16 threads per row, float4 per thread -> coalesced 256B rows. Non-temporal
// stores keep the L2 free for the 51 MB z table (and the index arrays).
// ---------------------------------------------------------------------------
__global__ void __launch_bounds__(256)
gather_kernel(const float* __restrict__ z,
              const int* __restrict__ idx_in,
              const int* __restrict__ idx_out,
              float* __restrict__ out, int n_out) {
    const long long t   = (long long)blockIdx.x * blockDim.x + threadIdx.x;
    const long long row = t >> 4;
    const int quad      = (int)(t & 15);
    if (row >= n_out) return;
    const int e    = idx_out[row];
    const int node = idx_in[e];
    const f32x4* src = (const f32x4*)(z + (size_t)node * DIM);
    f32x4 v = src[quad];
    f32x4* dst = (f32x4*)(out + (size_t)row * DIM) + quad;
    __builtin_nontemporal_store(v, dst);
}

// ---------------------------------------------------------------------------
// Fallback (only if workspace is too small): fused direct compute.
// ---------------------------------------------------------------------------
__global__ void __launch_bounds__(256)
fused_fallback_kernel(const float* __restrict__ values,
                      const float* __restrict__ w,
                      const int* __restrict__ idx_in,
                      const int* __restrict__ idx_out,
                      float* __restrict__ out, int n_out) {
    const long long t   = (long long)blockIdx.x * blockDim.x + threadIdx.x;
    const long long row = t >> 4;
    const int quad      = (int)(t & 15);
    if (row >= n_out) return;
    const int node = idx_in[idx_out[row]];
    const float* x = values + (size_t)node * DIM;
    float acc[4] = {0.f, 0.f, 0.f, 0.f};
    for (int i = 0; i < DIM; ++i) {
        const float xi = x[i];
#pragma unroll
        for (int j = 0; j < 4; ++j)
            acc[j] += w[(quad * 4 + j) * DIM + i] * xi;
    }
    f32x4 v;
    v.x = acc[0]; v.y = acc[1]; v.z = acc[2]; v.w = acc[3];
    __builtin_nontemporal_store(v, (f32x4*)(out + (size_t)row * DIM) + quad);
}

extern "C" void kernel_launch(void* const* d_in, const int* in_sizes, int n_in,
                              void* d_out, int out_size, void* d_ws, size_t ws_size,
                              hipStream_t stream) {
    const float* values  = (const float*)d_in[0];
    const float* w       = (const float*)d_in[1];
    const int*   idx_in  = (const int*)d_in[2];
    const int*   idx_out = (const int*)d_in[3];
    float*       out     = (float*)d_out;

    const int n_nodes    = in_sizes[0] / DIM;
    const int n_out_rows = in_sizes[3];
    const size_t z_bytes = (size_t)n_nodes * DIM * sizeof(float);

    if (ws_size >= z_bytes) {
        float* z = (float*)d_ws;
        const int n_tiles = (n_nodes + 15) / 16;
        const int blocks1 = (n_tiles + 7) / 8;     // 8 waves (tiles) per block
        linear_wmma_kernel<<<blocks1, 256, 0, stream>>>(values, w, z, n_nodes, n_tiles);

        const long long threads2 = (long long)n_out_rows * 16;
        const int blocks2 = (int)((threads2 + 255) / 256);
        gather_kernel<<<blocks2, 256, 0, stream>>>(z, idx_in, idx_out, out, n_out_rows);
    } else {
        const long long threads = (long long)n_out_rows * 16;
        const int blocks = (int)((threads + 255) / 256);
        fused_fallback_kernel<<<blocks, 256, 0, stream>>>(values, w, idx_in, idx_out,
                                                          out, n_out_rows);
    }
}